// CharacterLevelSpectral_27212912788373
// MI455X (gfx1250) — compile-verified
//
#include <hip/hip_runtime.h>
#include <hip/hip_bf16.h>
#include <math.h>

// ---------------------------------------------------------------------------
// out[b,s,d] = W[d] * y[b,s] + bias[d]
// where y[b,:] = Re( iFFT( lowpass( FFT( ids[b,:]/255 ) ) ) )
// (bias passes through the low-pass untouched: it is pure DC along s)
// ---------------------------------------------------------------------------

#define SEQ   8192
#define LOG2S 13
#define BATCH 16
#define DIM   256

typedef __attribute__((ext_vector_type(2))) float v2f;
typedef __attribute__((ext_vector_type(8))) float v8f;

// ---------------------------------------------------------------------------
// Kernel 1: one workgroup per batch row. Full length-8192 complex FFT in LDS
// (64 KB), mask frequencies [2048, 6144), inverse FFT, write real part / N.
// ---------------------------------------------------------------------------
__global__ void fft_lowpass_kernel(const int* __restrict__ ids,
                                   float* __restrict__ y) {
    extern __shared__ float2 buf[];   // 8192 complex = 64 KB dynamic LDS
    const int b   = blockIdx.x;
    const int tid = threadIdx.x;
    const int nt  = blockDim.x;       // 512 threads = 16 waves

    // Load normalized chars directly into bit-reversed positions.
    for (int i = tid; i < SEQ; i += nt) {
        float x = (float)ids[b * SEQ + i] * (1.0f / 255.0f);
        int r = (int)(__brev((unsigned)i) >> (32 - LOG2S));
        buf[r] = make_float2(x, 0.0f);
    }
    __syncthreads();

    // Forward FFT (DIT radix-2, numpy convention: e^{-2*pi*i*f*t/N}).
    for (int len = 2; len <= SEQ; len <<= 1) {
        const int   half = len >> 1;
        const float ang  = -6.283185307179586f / (float)len;
        for (int t = tid; t < (SEQ >> 1); t += nt) {
            int pos = t & (half - 1);
            int i0  = ((t ^ pos) << 1) | pos;  // (t-pos)*2 + pos
            int i1  = i0 + half;
            float s, c;
            sincosf(ang * (float)pos, &s, &c);
            float2 u = buf[i0];
            float2 w = buf[i1];
            float2 wv = make_float2(c * w.x - s * w.y, c * w.y + s * w.x);
            buf[i0] = make_float2(u.x + wv.x, u.y + wv.y);
            buf[i1] = make_float2(u.x - wv.x, u.y - wv.y);
        }
        __syncthreads();
    }

    // Low-pass: zero bins in [k, S-k), k = S/4 = 2048.
    for (int f = tid; f < SEQ; f += nt) {
        if (f >= (SEQ / 4) && f < (SEQ - SEQ / 4)) buf[f] = make_float2(0.0f, 0.0f);
    }
    __syncthreads();

    // In-place bit-reverse permutation (rev is an involution; owner of the
    // smaller index does the swap, the other thread only evaluates the guard).
    for (int i = tid; i < SEQ; i += nt) {
        int r = (int)(__brev((unsigned)i) >> (32 - LOG2S));
        if (r > i) { float2 tmp = buf[i]; buf[i] = buf[r]; buf[r] = tmp; }
    }
    __syncthreads();

    // Inverse FFT (positive twiddle sign), scale by 1/N at writeback.
    for (int len = 2; len <= SEQ; len <<= 1) {
        const int   half = len >> 1;
        const float ang  = 6.283185307179586f / (float)len;
        for (int t = tid; t < (SEQ >> 1); t += nt) {
            int pos = t & (half - 1);
            int i0  = ((t ^ pos) << 1) | pos;
            int i1  = i0 + half;
            float s, c;
            sincosf(ang * (float)pos, &s, &c);
            float2 u = buf[i0];
            float2 w = buf[i1];
            float2 wv = make_float2(c * w.x - s * w.y, c * w.y + s * w.x);
            buf[i0] = make_float2(u.x + wv.x, u.y + wv.y);
            buf[i1] = make_float2(u.x - wv.x, u.y - wv.y);
        }
        __syncthreads();
    }

    for (int s = tid; s < SEQ; s += nt) {
        y[b * SEQ + s] = buf[s].x * (1.0f / (float)SEQ);
    }
}

// ---------------------------------------------------------------------------
// Kernel 2: bandwidth-bound expansion. Each wave computes one 16(s) x 16(d)
// tile as a rank-1 WMMA:  D = A(16x4) * B(4x16) + C
//   A col 0 = y[s0..s0+15], B row 0 = W[d0..d0+15], C[m,n] = bias[d0+n].
// 16x16x4 f32 A/B layout: VGPR0 = {K=0 | lanes 0-15, K=2 | lanes 16-31},
// VGPR1 = {K=1, K=3}; only K=0 is nonzero here.
// ---------------------------------------------------------------------------
__global__ void expand_wmma_kernel(const float* __restrict__ y,
                                   const float* __restrict__ W,
                                   const float* __restrict__ bias,
                                   float* __restrict__ out) {
    const int lane   = threadIdx.x & 31;
    const int waveId = (int)((blockIdx.x * blockDim.x + threadIdx.x) >> 5);

    const int dTile = waveId & (DIM / 16 - 1);          // 16 tiles along d
    const int sTile = (waveId >> 4) & (SEQ / 16 - 1);   // 512 tiles along s
    const int b     = waveId >> 13;                     // batch row
    const int d0    = dTile << 4;
    const int s0    = sTile << 4;
    const int n     = lane & 15;                        // column within tile

    const bool lo = (lane < 16);
    float yv = lo ? y[b * SEQ + s0 + lane] : 0.0f;      // A: M=lane, K=0
    float wv = lo ? W[d0 + lane]           : 0.0f;      // B: N=lane, K=0

    v2f a;  a.x = yv;  a.y = 0.0f;                      // K=2 (hi lanes) zeroed
    v2f bm; bm.x = wv; bm.y = 0.0f;

    float bv = bias[d0 + n];
    v8f c;
#pragma unroll
    for (int v = 0; v < 8; ++v) c[v] = bv;

    v8f d = __builtin_amdgcn_wmma_f32_16x16x4_f32(
        /*neg_a=*/false, a, /*neg_b=*/false, bm,
        /*c_mod=*/(short)0, c, /*reuse_a=*/false, /*reuse_b=*/false);

    // D layout: lane holds N = lane%16; VGPR v holds M = v (lanes 0-15) or
    // M = v+8 (lanes 16-31). Each half-wave store is a contiguous 64B row run.
    float* outBase = out + ((size_t)b * SEQ + (size_t)s0) * DIM + d0;
    const int mOff = (lane >> 4) << 3;   // +8 rows for upper half-wave
#pragma unroll
    for (int v = 0; v < 8; ++v) {
        outBase[(size_t)(v + mOff) * DIM + n] = d[v];
    }
}

extern "C" void kernel_launch(void* const* d_in, const int* in_sizes, int n_in,
                              void* d_out, int out_size, void* d_ws, size_t ws_size,
                              hipStream_t stream) {
    const int*   ids  = (const int*)d_in[0];   // (B, S) char ids
    const float* Wmat = (const float*)d_in[1]; // (D, 1) -> flat D
    const float* bias = (const float*)d_in[2]; // (D,)
    float*       out  = (float*)d_out;         // (B, S, D) f32
    float*       y    = (float*)d_ws;          // (B, S) f32 scratch, 512 KB

    // Kernel 1: 16 workgroups (one per batch row), 512 threads, 64 KB LDS.
    fft_lowpass_kernel<<<BATCH, 512, SEQ * sizeof(float2), stream>>>(ids, y);

    // Kernel 2: one wave per 16x16 tile. Tiles = 16 * 512 * 16 = 131072 waves;
    // 8 waves (256 threads) per block -> 16384 blocks.
    const int totalWaves = BATCH * (SEQ / 16) * (DIM / 16);
    const int blockSize  = 256;
    const int wavesPerBlk = blockSize / 32;
    const int nBlocks    = totalWaves / wavesPerBlk;
    expand_wmma_kernel<<<nBlocks, blockSize, 0, stream>>>(y, Wmat, bias, out);
}